// GraphDecoder_16423954940120
// MI455X (gfx1250) — compile-verified
//
#include <hip/hip_runtime.h>
#include <hip/hip_bf16.h>

// ---------------------------------------------------------------------------
// GraphDecoder fused pipeline for MI455X (gfx1250, wave32, WMMA f32 16x16x4).
//
// Algebraic fusion: edge_in @ W_e1 is decomposed through the concat structure,
// so the O(B*E*576*512) GEMM (185 GFLOP, 722MB intermediate) collapses to
// small precomputed GEMMs + a fused per-edge ReLU + K=512->5 WMMA GEMM
// (~3.6 GFLOP). All intermediates (~57 MB) are L2 resident (192 MB L2);
// HBM traffic ~= outputs only.
//
// All ragged dims (K=131 input, N=50 size-head) are zero-padded in workspace
// so WMMA inner loops have NO per-element guards and every load is in-bounds:
// unconditional b64 loads, s_clause batching, staggered s_wait_loadcnt.
// ---------------------------------------------------------------------------

typedef float v2f __attribute__((ext_vector_type(2)));
typedef float v8f __attribute__((ext_vector_type(8)));

#define B_SZ   256
#define L_DIM  128
#define P_DIM  3
#define KINP   132               // 131 padded to multiple of 4
#define H_DIM  512
#define H2_DIM 256
#define N_NODES 50
#define NPAD   64                // 50 padded to multiple of 16
#define D_DIM  32
#define ND_DIM (N_NODES * D_DIM) // 1600
#define EDIMP1 5                 // EDIM + 1
#define E_CNT  1225              // N*(N-1)/2

__device__ __forceinline__ v8f wmma4(v2f a, v2f b, v8f c) {
  // V_WMMA_F32_16X16X4_F32 : D[16x16] += A[16x4] * B[4x16], exact fp32 path
  return __builtin_amdgcn_wmma_f32_16x16x4_f32(
      /*neg_a=*/false, a, /*neg_b=*/false, b,
      /*c_mod=*/(short)0, c, /*reuse_a=*/false, /*reuse_b=*/false);
}

// ---------------------------------------------------------------------------
// C[M,N] = act(A[M,K] @ W[K,N] + bias) ; one 16x16 tile per wave.
// REQUIRES: M%16==0, K%4==0, N covered by wave-uniform tile guard, and both
// A [M,K] and W [K,N] fully in-bounds (ragged operands pre-padded in ws).
// Fragment layouts per CDNA5 ISA 7.12.2:
//   A 16x4 : lane&15 = row M, (lane>>4)*2 (+0/+1) = K   -> one b64 load
//   B 4x16 : lane&15 = col N, same K mapping            -> two b32 loads
//   C/D    : vgpr v -> row = v + 8*(lane>=16), col = lane&15
// ---------------------------------------------------------------------------
__global__ void gemm16_kernel(const float* __restrict__ A,
                              const float* __restrict__ W,
                              const float* __restrict__ bias,
                              float* __restrict__ C,
                              int M, int Ncols, int K, int do_relu) {
  const int lane = threadIdx.x & 31;
  const int wave = threadIdx.x >> 5;
  const int tm = blockIdx.x;
  const int tn = blockIdx.y * 8 + wave;
  if (tn * 16 >= Ncols) return;              // wave-uniform exit (EXEC stays full)

  const int m16  = lane & 15;
  const int mrow = tm * 16 + m16;
  const int ncol = tn * 16 + m16;
  const int kk   = (lane >> 4) * 2;

  const float* Ap = A + (size_t)mrow * K + kk;   // 8B aligned (kk even, K%4==0)
  const float* Wp = W + (size_t)kk * Ncols + ncol;

  v8f acc = {0.f, 0.f, 0.f, 0.f, 0.f, 0.f, 0.f, 0.f};
#pragma unroll 4
  for (int kb = 0; kb < K; kb += 4) {
    const v2f a = *(const v2f*)(Ap + kb);
    v2f b;
    b.x = Wp[(size_t)kb * Ncols];
    b.y = Wp[(size_t)kb * Ncols + Ncols];
    acc = wmma4(a, b, acc);
  }
#pragma unroll
  for (int v = 0; v < 8; ++v) {
    const int row = tm * 16 + v + ((lane >> 4) << 3);
    if (row < M && ncol < Ncols) {
      float val = acc[v];
      if (bias) val += bias[ncol];
      if (do_relu) val = fmaxf(val, 0.f);
      C[(size_t)row * Ncols + ncol] = val;
    }
  }
}

// peW[e,:] = (0.5*(pos_emb[ii[e]] + pos_emb[jj[e]])) @ W1h   [E=1225, 512]
__global__ void pew_kernel(const float* __restrict__ pos_emb,
                           const float* __restrict__ W1h,
                           const int* __restrict__ ii, const int* __restrict__ jj,
                           float* __restrict__ peW) {
  const int lane = threadIdx.x & 31;
  const int wave = threadIdx.x >> 5;
  const int tm = blockIdx.x;                  // 0..76
  const int tn = blockIdx.y * 8 + wave;       // 0..31
  const int m16 = lane & 15;
  const int e   = tm * 16 + m16;
  const int ec  = (e < E_CNT) ? e : (E_CNT - 1);   // clamp: loads stay unconditional
  const int kk   = (lane >> 4) * 2;
  const int ncol = tn * 16 + m16;

  const float* pi = pos_emb + (size_t)ii[ec] * H_DIM + kk;
  const float* pj = pos_emb + (size_t)jj[ec] * H_DIM + kk;
  const float* Wp = W1h + (size_t)kk * H_DIM + ncol;

  v8f acc = {0.f, 0.f, 0.f, 0.f, 0.f, 0.f, 0.f, 0.f};
#pragma unroll 4
  for (int kb = 0; kb < H_DIM; kb += 4) {
    const v2f vi = *(const v2f*)(pi + kb);
    const v2f vj = *(const v2f*)(pj + kb);
    v2f a, b;
    a.x = 0.5f * (vi.x + vj.x);
    a.y = 0.5f * (vi.y + vj.y);
    b.x = Wp[(size_t)kb * H_DIM];
    b.y = Wp[(size_t)kb * H_DIM + H_DIM];
    acc = wmma4(a, b, acc);
  }
#pragma unroll
  for (int v = 0; v < 8; ++v) {
    const int ee = tm * 16 + v + ((lane >> 4) << 3);
    if (ee < E_CNT) peW[(size_t)ee * H_DIM + ncol] = acc[v];
  }
}

// Fused edge evaluator:
// edge_logits[b,e,:5] = ReLU(hb[b] + peW[e] + nlWa[b,ii[e]] + nlWb[b,jj[e]]) @ W_e2 + b_e2
// 16 edges x 16 outs (5 live) per wave, K=512 via WMMA.
// W_e2 zero-padded in LDS with pair-interleaved layout [k/2][n][2] so the
// B fragment is a single conflict-free ds_load_b64.
__global__ void edge_kernel(const float* __restrict__ hb,
                            const float* __restrict__ peW,
                            const float* __restrict__ nla,
                            const float* __restrict__ nlb,
                            const float* __restrict__ W_e2,
                            const float* __restrict__ b_e2,
                            const int* __restrict__ ii, const int* __restrict__ jj,
                            float* __restrict__ out_edge) {
  __shared__ float w2[H_DIM * 16];            // 32 KB
  for (int t = threadIdx.x; t < H_DIM * 16; t += blockDim.x) {
    const int p = t >> 5;            // K pair index 0..255
    const int n = (t >> 1) & 15;     // output col
    const int r = t & 1;             // K within pair
    const int k = 2 * p + r;
    w2[t] = (n < EDIMP1) ? W_e2[k * EDIMP1 + n] : 0.f;
  }
  __syncthreads();

  const int lane = threadIdx.x & 31;
  const int wave = threadIdx.x >> 5;
  const int b    = blockIdx.y;
  const int tile = blockIdx.x * 8 + wave;     // 0..79 (77 live)
  const int m16  = lane & 15;
  const int e    = tile * 16 + m16;
  const int ec   = (e < E_CNT) ? e : (E_CNT - 1);
  const int kk   = (lane >> 4) * 2;

  const float* hbp = hb  + (size_t)b * H_DIM + kk;
  const float* pep = peW + (size_t)ec * H_DIM + kk;
  const float* nap = nla + ((size_t)b * N_NODES + ii[ec]) * H_DIM + kk;
  const float* nbp = nlb + ((size_t)b * N_NODES + jj[ec]) * H_DIM + kk;
  const float* w2p = w2 + (size_t)(kk >> 1) * 32 + m16 * 2;  // pair-interleaved

  v8f acc = {0.f, 0.f, 0.f, 0.f, 0.f, 0.f, 0.f, 0.f};
#pragma unroll 4
  for (int kb = 0; kb < H_DIM; kb += 4) {
    const v2f vh = *(const v2f*)(hbp + kb);
    const v2f vp = *(const v2f*)(pep + kb);
    const v2f va = *(const v2f*)(nap + kb);
    const v2f vb = *(const v2f*)(nbp + kb);
    v2f a;
    a.x = fmaxf(vh.x + vp.x + va.x + vb.x, 0.f);
    a.y = fmaxf(vh.y + vp.y + va.y + vb.y, 0.f);
    const v2f bf = *(const v2f*)(w2p + (size_t)(kb >> 1) * 32);
    acc = wmma4(a, bf, acc);
  }
#pragma unroll
  for (int v = 0; v < 8; ++v) {
    const int ee = tile * 16 + v + ((lane >> 4) << 3);
    if (ee < E_CNT && m16 < EDIMP1)
      out_edge[((size_t)b * E_CNT + ee) * EDIMP1 + m16] = acc[v] + b_e2[m16];
  }
}

// zin[b, 0:132] = concat(z[b], target_properties[b], 0-pad)
__global__ void pack_kernel(const float* __restrict__ z, const float* __restrict__ tp,
                            float* __restrict__ zin) {
  const int idx = blockIdx.x * blockDim.x + threadIdx.x;
  if (idx >= B_SZ * KINP) return;
  const int r = idx / KINP, c = idx % KINP;
  float v = 0.f;
  if (c < L_DIM)               v = z[r * L_DIM + c];
  else if (c < L_DIM + P_DIM)  v = tp[r * P_DIM + (c - L_DIM)];
  zin[idx] = v;
}

// W_in [131,512] -> w_inp [132,512] zero-padded last row (keeps B-loads in-bounds)
__global__ void pad_win_kernel(const float* __restrict__ W_in, float* __restrict__ w_inp) {
  const int idx = blockIdx.x * blockDim.x + threadIdx.x;
  if (idx >= KINP * H_DIM) return;
  const int k = idx / H_DIM;
  w_inp[idx] = (k < L_DIM + P_DIM) ? W_in[idx] : 0.f;
}

// W_s2 [256,50] -> w_s2p [256,64] zero-padded columns
__global__ void pad_ws2_kernel(const float* __restrict__ W_s2, float* __restrict__ w_s2p) {
  const int idx = blockIdx.x * blockDim.x + threadIdx.x;
  if (idx >= H2_DIM * NPAD) return;
  const int k = idx / NPAD, n = idx % NPAD;
  w_s2p[idx] = (n < N_NODES) ? W_s2[k * N_NODES + n] : 0.f;
}

// triu(k=1) index pairs, row-major order; also emitted to d_out as int32.
__global__ void edge_idx_kernel(int* __restrict__ ii, int* __restrict__ jj,
                                int* __restrict__ out_idx) {
  const int e = blockIdx.x * blockDim.x + threadIdx.x;
  if (e >= E_CNT) return;
  int rem = e, i = 0;
  while (rem >= N_NODES - 1 - i) { rem -= N_NODES - 1 - i; ++i; }
  const int j = i + 1 + rem;
  ii[e] = i; jj[e] = j;
  out_idx[e * 2]     = i;
  out_idx[e * 2 + 1] = j;
}

// row-wise softmax over 50 live cols of stride-64 slog, bias folded in here
__global__ void softmax_kernel(const float* __restrict__ slog, const float* __restrict__ bias,
                               float* __restrict__ out) {
  const int r = blockIdx.x * blockDim.x + threadIdx.x;
  if (r >= B_SZ) return;
  const float* p = slog + (size_t)r * NPAD;
  float mx = -3.4e38f;
  for (int c = 0; c < N_NODES; ++c) mx = fmaxf(mx, p[c] + bias[c]);
  float s = 0.f;
  for (int c = 0; c < N_NODES; ++c) s += expf(p[c] + bias[c] - mx);
  const float inv = 1.f / s;
  for (int c = 0; c < N_NODES; ++c) out[(size_t)r * N_NODES + c] = expf(p[c] + bias[c] - mx) * inv;
}

// ---------------------------------------------------------------------------

extern "C" void kernel_launch(void* const* d_in, const int* in_sizes, int n_in,
                              void* d_out, int out_size, void* d_ws, size_t ws_size,
                              hipStream_t stream) {
  const float* z    = (const float*)d_in[0];
  const float* tp   = (const float*)d_in[1];
  const float* W_in = (const float*)d_in[2];
  const float* b_in = (const float*)d_in[3];
  const float* W_s1 = (const float*)d_in[4];
  const float* b_s1 = (const float*)d_in[5];
  const float* W_s2 = (const float*)d_in[6];
  const float* b_s2 = (const float*)d_in[7];
  const float* W_n1 = (const float*)d_in[8];
  const float* b_n1 = (const float*)d_in[9];
  const float* W_n2 = (const float*)d_in[10];
  const float* b_n2 = (const float*)d_in[11];
  const float* W_e1 = (const float*)d_in[12];
  const float* b_e1 = (const float*)d_in[13];
  const float* W_e2 = (const float*)d_in[14];
  const float* b_e2 = (const float*)d_in[15];
  const float* pos_emb = (const float*)d_in[16];

  // d_out layout (flat, return order): node_logits | edge_logits | size_probs | edge_indices
  float* out_node  = (float*)d_out;                                  // 256*50*32 = 409600
  float* out_edge  = out_node + (size_t)B_SZ * ND_DIM;               // 256*1225*5 = 1568000
  float* out_sizep = out_edge + (size_t)B_SZ * E_CNT * EDIMP1;       // 256*50    = 12800
  int*   out_idx   = (int*)(out_sizep + (size_t)B_SZ * N_NODES);     // 1225*2 int32

  // workspace layout (floats); requires ~57.5 MB of ws
  float* ws = (float*)d_ws;
  float* zin   = ws;                                    // 256*132 = 33792
  float* w_inp = zin   + (size_t)B_SZ * KINP;           // 132*512 = 67584
  float* h     = w_inp + (size_t)KINP * H_DIM;          // 131072
  float* s1    = h     + (size_t)B_SZ * H_DIM;          // 65536
  float* slog  = s1    + (size_t)B_SZ * H2_DIM;         // 256*64 = 16384 (padded)
  float* wsp2  = slog  + (size_t)B_SZ * NPAD;           // 256*64 = 16384 (padded W_s2)
  float* n1    = wsp2  + (size_t)H2_DIM * NPAD;         // 131072
  float* hb    = n1    + (size_t)B_SZ * H_DIM;          // 131072
  float* peW   = hb    + (size_t)B_SZ * H_DIM;          // 627200
  float* nla   = peW   + (size_t)E_CNT * H_DIM;         // 6553600
  float* nlb   = nla   + (size_t)B_SZ * N_NODES * H_DIM; // 6553600
  int*   ii    = (int*)(nlb + (size_t)B_SZ * N_NODES * H_DIM);
  int*   jj    = ii + E_CNT;

  const float* W1h = W_e1;                    // rows 0..511   of [576,512]
  const float* W1a = W_e1 + 512 * H_DIM;      // rows 512..543
  const float* W1b = W_e1 + 544 * H_DIM;      // rows 544..575

  // 0) edge indices (also final output) + padded operands
  edge_idx_kernel<<<dim3((E_CNT + 255) / 256), dim3(256), 0, stream>>>(ii, jj, out_idx);
  pack_kernel<<<dim3((B_SZ * KINP + 255) / 256), dim3(256), 0, stream>>>(z, tp, zin);
  pad_win_kernel<<<dim3((KINP * H_DIM + 255) / 256), dim3(256), 0, stream>>>(W_in, w_inp);
  pad_ws2_kernel<<<dim3((H2_DIM * NPAD + 255) / 256), dim3(256), 0, stream>>>(W_s2, wsp2);
  // 1) h = relu(zin @ w_inp + b_in)                      [256,512], K=132 (both padded)
  gemm16_kernel<<<dim3(16, 4), dim3(256), 0, stream>>>(zin, w_inp, b_in, h, B_SZ, H_DIM, KINP, 1);
  // 2) s1 = relu(h @ W_s1 + b_s1)                        [256,256], K=512
  gemm16_kernel<<<dim3(16, 2), dim3(256), 0, stream>>>(h, W_s1, b_s1, s1, B_SZ, H2_DIM, H_DIM, 1);
  // 3) slog = s1 @ w_s2p (padded N=64, bias in softmax)  [256,64],  K=256
  gemm16_kernel<<<dim3(16, 1), dim3(256), 0, stream>>>(s1, wsp2, nullptr, slog, B_SZ, NPAD, H2_DIM, 0);
  // 4) size_probs = softmax(slog + b_s2)
  softmax_kernel<<<dim3(1), dim3(256), 0, stream>>>(slog, b_s2, out_sizep);
  // 5) n1 = relu(h @ W_n1 + b_n1)                        [256,512], K=512
  gemm16_kernel<<<dim3(16, 4), dim3(256), 0, stream>>>(h, W_n1, b_n1, n1, B_SZ, H_DIM, H_DIM, 1);
  // 6) node_logits = n1 @ W_n2 + b_n2                    [256,1600], K=512 (final output)
  gemm16_kernel<<<dim3(16, 13), dim3(256), 0, stream>>>(n1, W_n2, b_n2, out_node, B_SZ, ND_DIM, H_DIM, 0);
  // 7) hb = h @ W1h + b_e1                               [256,512], K=512
  gemm16_kernel<<<dim3(16, 4), dim3(256), 0, stream>>>(h, W1h, b_e1, hb, B_SZ, H_DIM, H_DIM, 0);
  // 8) nlWa = node_logits @ W1a ; nlWb = node_logits @ W1b   [12800,512], K=32
  gemm16_kernel<<<dim3(800, 4), dim3(256), 0, stream>>>(out_node, W1a, nullptr, nla,
                                                        B_SZ * N_NODES, H_DIM, D_DIM, 0);
  gemm16_kernel<<<dim3(800, 4), dim3(256), 0, stream>>>(out_node, W1b, nullptr, nlb,
                                                        B_SZ * N_NODES, H_DIM, D_DIM, 0);
  // 9) peW = (0.5*(pe[i]+pe[j])) @ W1h                   [1225,512], K=512
  pew_kernel<<<dim3(77, 4), dim3(256), 0, stream>>>(pos_emb, W1h, ii, jj, peW);
  // 10) fused edge MLP: ReLU(sum of 4 precomputed terms) @ W_e2 + b_e2
  edge_kernel<<<dim3(10, B_SZ), dim3(256), 0, stream>>>(hb, peW, nla, nlb, W_e2, b_e2,
                                                        ii, jj, out_edge);
  (void)in_sizes; (void)n_in; (void)out_size; (void)ws_size;
}